// PhysicalCircleLayer_61091614819011
// MI455X (gfx1250) — compile-verified
//
#include <hip/hip_runtime.h>
#include <hip/hip_bf16.h>

// ---------------------------------------------------------------------------
// PhysicalCircleLayer on gfx1250 (MI455X)
// Bandwidth-bound segmented-min over 164MB of seg_maps (~7us HBM floor).
// CDNA5-specific paths used:
//   * global_load_async_to_lds_b128 double-buffered streaming (ASYNCcnt)
//   * s_wait_asynccnt split counters
//   * ds_min_u32 non-returning LDS atomics for the segmented min
// No matmul structure exists in this op, so WMMA is not applicable.
// ---------------------------------------------------------------------------

#define AS1 __attribute__((address_space(1)))
#define AS3 __attribute__((address_space(3)))

typedef int v4i __attribute__((ext_vector_type(4)));

#define SDIM     100
#define NPIX     10000          // SDIM*SDIM
#define BLK      256            // 8 wave32
#define CHUNK    1024           // floats per double-buffer stage (4KB)
#define NCHUNK   10             // ceil(10000/1024)
#define ACC_STR  25             // 24 accumulators + 1 pad -> 32 distinct banks/wave
#define INF_F    1000000000.0f
#define MU_F     1e-8f

// builtin wants: (v4i addrspace(1)*, v4i addrspace(3)*, imm offset, imm cpol)
__device__ __forceinline__ void async_copy_b128(const float* gsrc, float* ldst) {
    __builtin_amdgcn_global_load_async_to_lds_b128(
        (AS1 v4i*)(v4i*)(void*)gsrc,
        (AS3 v4i*)(v4i*)(void*)ldst,
        0, 0);
}

__global__ __launch_bounds__(BLK)
void PhysicalCircleLayer_61091614819011_kernel(
    const float* __restrict__ seg_maps,   // (B, 100, 100)
    const float* __restrict__ paras,      // (B, 4)  [W0, W1, b0, b1]
    const float* __restrict__ traj,       // (B, 10, 2)
    const float* __restrict__ cpos,       // (B, 1, 2)
    float* __restrict__ out)              // (B, 16, 3)
{
    __shared__ float    stage[2][CHUNK];          // 8 KB  async landing pads
    __shared__ unsigned accum[BLK * ACC_STR];     // 25.6 KB per-thread min slots
    __shared__ unsigned scratch[8 * 24];          // stage-1 reduce
    __shared__ float    fin[24];                  // final per-(bin,bucket) mins

    const int b   = blockIdx.x;
    const int tid = threadIdx.x;
    const float* gmap = seg_maps + (size_t)b * NPIX;

    // --- per-batch uniforms ------------------------------------------------
    const float W0  = paras[b * 4 + 0], W1  = paras[b * 4 + 1];
    const float bv0 = paras[b * 4 + 2], bv1 = paras[b * 4 + 3];
    const float iW0 = 1.0f / W0,        iW1 = 1.0f / W1;
    const float cp0 = cpos[b * 2 + 0],  cp1 = cpos[b * 2 + 1];
    // moving_len: current_pos cancels in obs[-1]-obs[0]
    const float dxm = traj[b * 20 + 18] - traj[b * 20 + 0];
    const float dym = traj[b * 20 + 19] - traj[b * 20 + 1];
    const float ml  = sqrtf(dxm * dxm + dym * dym);
    const float r0 = 2.0f * ml, r1 = 5.0f * ml, r2 = 10.0f * ml;

    // --- init private accumulators (positive floats -> u32 order-preserving)
    const unsigned infb = __float_as_uint(INF_F);
#pragma unroll
    for (int j = 0; j < 24; ++j) accum[tid * ACC_STR + j] = infb;

    const int t4 = tid * 4;

    // --- prime the async pipeline: chunk 0 ---------------------------------
    async_copy_b128(gmap + t4, &stage[0][t4]);    // chunk 0 always in range

    for (int c = 0; c < NCHUNK; ++c) {
        const int nb = c + 1;
        if (nb < NCHUNK) {
            // Unconditional issue per wave: clamp OOB lanes so every wave's
            // ASYNCcnt increments uniformly (clamped lanes' stage slots are
            // never consumed — their compute guard fails below).
            int off = nb * CHUNK + t4;
            off = (off > NPIX - 4) ? (NPIX - 4) : off;
            async_copy_b128(gmap + off, &stage[nb & 1][t4]);
            __builtin_amdgcn_s_wait_asynccnt(1);    // chunk c retired, c+1 in flight
        } else {
            __builtin_amdgcn_s_wait_asynccnt(0);    // last chunk retired
        }

        const int i0 = c * CHUNK + t4;
        if (i0 < NPIX) {
            // each lane reads only bytes its own wave's async op wrote
            const float4 mv = *(const float4*)&stage[c & 1][t4];
            const float mvals[4] = {mv.x, mv.y, mv.z, mv.w};

            // i0 and 100 are both multiples of 4 -> y0 in {0,4,...,96},
            // so the 4-pixel group never wraps a row.
            const int x  = i0 / SDIM;
            const int y0 = i0 - x * SDIM;
            const float d0   = ((float)x - bv0) * iW0 - cp0;   // dvec[...,0]
            const float d0sq = d0 * d0;
            const float av   = fabsf(d0);
            const bool  vneg = d0 < 0.0f;

#pragma unroll
            for (int j = 0; j < 4; ++j) {
                const float d1   = ((float)(y0 + j) - bv1) * iW1 - cp1; // dvec[...,1]
                const float dist = sqrtf(fmaf(d1, d1, d0sq));

                // octant of atan2(d0, d1) mod 2pi in [k*pi/4,(k+1)*pi/4)
                // derived from signs + |d0| vs |d1| (no atan2/fmod)
                const float au   = fabsf(d1);
                const bool  uneg = d1 < 0.0f;
                const int bin = (vneg ? 4 : 0)
                              + ((uneg != vneg) ? 2 : 0)
                              + ((uneg == vneg) ? (av >= au ? 1 : 0)
                                                : (av <= au ? 1 : 0));

                const float mj   = mvals[j];
                const float bsel = (mj <= 0.05f) ? INF_F
                                                 : (dist + MU_F) / (mj + MU_F);
                // smallest radius bucket containing this pixel (r0<r1<r2)
                const int bucket = (dist <= r0) ? 0 :
                                   (dist <= r1) ? 1 :
                                   (dist <= r2) ? 2 : 3;
                if (bucket < 3) {
                    // ds_min_u32, private slot: no contention, 1 LDS op/pixel
                    atomicMin(&accum[tid * ACC_STR + bin * 3 + bucket],
                              __float_as_uint(bsel));
                }
            }
        }
    }

    __syncthreads();

    // --- stage-1 reduce: 8 groups x 24 slots (192 threads) -----------------
    if (tid < 192) {
        const int g = tid / 24, idx = tid - g * 24;
        unsigned mn = infb;
        const int base = g * 32;
        for (int s = 0; s < 32; ++s) {
            const unsigned v = accum[(base + s) * ACC_STR + idx];
            mn = (v < mn) ? v : mn;
        }
        scratch[g * 24 + idx] = mn;
    }
    __syncthreads();

    // --- stage-2 reduce: fold 8 groups -------------------------------------
    if (tid < 24) {
        unsigned mn = infb;
        for (int g = 0; g < 8; ++g) {
            const unsigned v = scratch[g * 24 + tid];
            mn = (v < mn) ? v : mn;
        }
        fin[tid] = __uint_as_float(mn);
    }
    __syncthreads();

    // --- prefix-min over radius buckets + pad to 16 partitions -------------
    if (tid < 48) {
        float o = 0.0f;
        if (tid < 24) {
            const int p = tid / 3, k = tid - p * 3;
            float m = fin[p * 3];                       // bucket 0
            if (k >= 1) m = fminf(m, fin[p * 3 + 1]);   // + bucket 1
            if (k >= 2) m = fminf(m, fin[p * 3 + 2]);   // + bucket 2
            o = (m < INF_F) ? m : 0.0f;
        }
        out[(size_t)b * 48 + tid] = o;                  // (p,k) = tid/3, tid%3
    }
}

extern "C" void kernel_launch(void* const* d_in, const int* in_sizes, int n_in,
                              void* d_out, int out_size, void* d_ws, size_t ws_size,
                              hipStream_t stream) {
    const float* seg_maps = (const float*)d_in[0];   // (B,100,100)
    const float* paras    = (const float*)d_in[1];   // (B,4)
    const float* traj     = (const float*)d_in[2];   // (B,10,2)
    const float* cpos     = (const float*)d_in[3];   // (B,1,2)
    // d_in[4] = map_pos_pixel: recomputed analytically on device.
    (void)n_in; (void)d_ws; (void)ws_size; (void)out_size;

    const int B = in_sizes[1] / 4;
    PhysicalCircleLayer_61091614819011_kernel<<<B, BLK, 0, stream>>>(
        seg_maps, paras, traj, cpos, (float*)d_out);
}